// polarrcnn_target_88227218195177
// MI455X (gfx1250) — compile-verified
//
#include <hip/hip_runtime.h>

// Ext-vector types: required for __builtin_nontemporal_load (b128 NT loads)
// and for the WMMA builtin operand shapes.
typedef float v2f __attribute__((ext_vector_type(2)));
typedef float v4f __attribute__((ext_vector_type(4)));
typedef float v8f __attribute__((ext_vector_type(8)));

__global__ void polarrcnn_zero_out(float* out) {
    if (threadIdx.x == 0 && blockIdx.x == 0) out[0] = 0.0f;
}

// Streaming masked reduction:
//   total = sum_i [cls[i] >= 0.5] * (cls[i] + ep[2i] + ep[2i+1])
// HBM-bound: 96 MB single pass -> use non-temporal b128 loads.
// Wave reduction done with V_WMMA_F32_16X16X4_F32 (A = per-lane {a0,a1} = 16x4
// fragment, B = ones => D rows hold row-sums; 8 VGPR sums + shfl_xor(16) gives
// the wave total in every lane).
__global__ __launch_bounds__(256) void polarrcnn_masked_sum(
    const float* __restrict__ cls,
    const float* __restrict__ ep,
    float* __restrict__ out,
    int n)
{
    const int n4 = n >> 2;
    const v4f* __restrict__ cls4 = (const v4f*)cls;
    const v4f* __restrict__ ep4  = (const v4f*)ep;

    float a0 = 0.0f, a1 = 0.0f;
    const int stride = gridDim.x * blockDim.x;
    const int tid    = blockIdx.x * blockDim.x + threadIdx.x;

    // Main vectorized stream: 16B cls + 32B ep per iteration, NT-hinted.
    for (int i = tid; i < n4; i += stride) {
        v4f c  = __builtin_nontemporal_load(cls4 + i);
        v4f e0 = __builtin_nontemporal_load(ep4 + 2 * i);
        v4f e1 = __builtin_nontemporal_load(ep4 + 2 * i + 1);
        a0 += (c.x >= 0.5f) ? (c.x + e0.x + e0.y) : 0.0f;
        a1 += (c.y >= 0.5f) ? (c.y + e0.z + e0.w) : 0.0f;
        a0 += (c.z >= 0.5f) ? (c.z + e1.x + e1.y) : 0.0f;
        a1 += (c.w >= 0.5f) ? (c.w + e1.z + e1.w) : 0.0f;
    }
    // Scalar tail (N = 8M is divisible by 4, but stay generic).
    for (int i = n4 * 4 + tid; i < n; i += stride) {
        float c = cls[i];
        if (c >= 0.5f) a0 += c + ep[2 * i] + ep[2 * i + 1];
    }

    // ---- Wave32 reduction via WMMA: D = A(16x4) * ones(4x16) + 0 ----
    // A fragment layout (32-bit A, 16x4): lanes 0-15 hold K=0/1 in VGPR0/1,
    // lanes 16-31 hold K=2/3. Per-lane {a0,a1} fills the whole matrix.
    v2f A; A.x = a0;   A.y = a1;
    v2f B; B.x = 1.0f; B.y = 1.0f;   // all-ones B (every element == 1.0)
    v8f C = {};
    v8f D = __builtin_amdgcn_wmma_f32_16x16x4_f32(
        /*neg_a=*/false, A, /*neg_b=*/false, B,
        /*c_mod=*/(short)0, C, /*reuse_a=*/false, /*reuse_b=*/false);

    // Lane n<16: VGPRs 0..7 of D = rows 0..7 sums (col n); lanes 16-31: rows 8..15.
    float s = D[0] + D[1] + D[2] + D[3] + D[4] + D[5] + D[6] + D[7];
    s += __shfl_xor(s, 16, 32);   // add complementary half -> wave total

    // ---- Block reduction: 8 waves -> LDS -> 1 atomic per block ----
    __shared__ float lsum[8];
    const int lane = threadIdx.x & 31;
    const int wave = threadIdx.x >> 5;
    if (lane == 0) lsum[wave] = s;
    __syncthreads();
    if (threadIdx.x == 0) {
        float t = 0.0f;
#pragma unroll
        for (int j = 0; j < 8; ++j) t += lsum[j];
        atomicAdd(out, t);
    }
}

extern "C" void kernel_launch(void* const* d_in, const int* in_sizes, int n_in,
                              void* d_out, int out_size, void* d_ws, size_t ws_size,
                              hipStream_t stream) {
    const float* cls = (const float*)d_in[0];   // [N] f32
    const float* ep  = (const float*)d_in[1];   // [N,2] f32
    float* out = (float*)d_out;                 // scalar f32
    const int n = in_sizes[0];

    polarrcnn_zero_out<<<1, 64, 0, stream>>>(out);

    int n4 = n >> 2;
    int blocks = (n4 + 255) / 256;
    if (blocks > 4096) blocks = 4096;           // ~1M threads, grid-stride covers rest
    if (blocks < 1) blocks = 1;
    polarrcnn_masked_sum<<<blocks, 256, 0, stream>>>(cls, ep, out, n);
}